// GAT_1082331759085
// MI455X (gfx1250) — compile-verified
//
#include <hip/hip_runtime.h>
#include <math.h>

// ---------------------------------------------------------------------------
// GAT (2-layer) for MI455X / gfx1250, fp32.
//   Layer 1: H=8, C=8  (concat -> 64), Layer 2: H=1, C=64 (mean of 1 head).
// GEMMs use V_WMMA_F32_16X16X4_F32 (fp32-exact matrix cores), with W staged
// in LDS in a K-pair-interleaved layout so each B fragment is ONE ds_load_b64.
// Edge phase: init -> segment-max -> segment-expsum -> weighted scatter-add,
// all L2-resident on MI455X (working set << 192MB L2).
// ---------------------------------------------------------------------------

typedef __attribute__((ext_vector_type(2))) float v2f;
typedef __attribute__((ext_vector_type(8))) float v8f;

__device__ __forceinline__ float lrelu02(float v) {
    return v > 0.0f ? v : 0.2f * v;
}

// Monotone-mapping float atomic max (works with emax initialized to -inf).
__device__ __forceinline__ void atomicMaxF(float* addr, float v) {
    int iv = __float_as_int(v);
    if (iv >= 0) atomicMax((int*)addr, iv);
    else         atomicMin((unsigned int*)addr, (unsigned int)iv);
}

__global__ void fill_kernel(float* __restrict__ p, size_t n, float v) {
    size_t i = (size_t)blockIdx.x * blockDim.x + threadIdx.x;
    if (i < n) p[i] = v;
}

// ---------------------------------------------------------------------------
// Y[M,64] = X[M,K] @ W[K,64]  (K = 128 or 64), fp32 WMMA 16x16x4.
// 256 threads = 8 wave32s; each wave computes a 16x64 strip.
//
// LDS holds W interleaved by K-pairs:
//   sW[(k>>1)*128 + 2*n + (k&1)] = W[k][n]
// so the B fragment {W[k][n], W[k+1][n]} is a single aligned 8B LDS word.
//
// A 16x4 layout : lanes 0-15 hold M=0..15 {K0,K1}; lanes 16-31 {K2,K3}
// B 4x16 layout : lanes 0-15 hold N=0..15 {K0,K1}; lanes 16-31 {K2,K3}
// C 16x16 layout: VGPR j -> (M=j, N=lane) | (M=j+8, N=lane-16)
// ---------------------------------------------------------------------------
__global__ __launch_bounds__(256) void gemm_wmma_kernel(
    const float* __restrict__ X, const float* __restrict__ W,
    float* __restrict__ Y, int M, int K)
{
    __shared__ float sW[128 * 64];
    const int nW = K * 64;
    for (int i = threadIdx.x; i < nW; i += 256) {
        int k = i >> 6, n = i & 63;
        sW[(k >> 1) * 128 + 2 * n + (k & 1)] = W[i];
    }
    __syncthreads();

    const int wave = threadIdx.x >> 5;
    const int lane = threadIdx.x & 31;
    const int m0   = (blockIdx.x * 8 + wave) * 16;
    if (m0 >= M) return;

    const int half = lane >> 4;   // 0: K0/K1 side, 1: K2/K3 side
    const int r    = lane & 15;

    v8f c0 = {}, c1 = {}, c2 = {}, c3 = {};

    const float* arow = X + (size_t)(m0 + r) * K + 2 * half;

    for (int k0 = 0; k0 < K; k0 += 4) {
        v2f a = *(const v2f*)(arow + k0);              // {K0,K1} or {K2,K3}

        // pair index: (k0 + 2*half)/2 = k0/2 + half
        const float* wp = &sW[((k0 >> 1) + half) * 128 + 2 * r];
        v2f b0 = *(const v2f*)(wp +  0);               // cols  0..15
        v2f b1 = *(const v2f*)(wp + 32);               // cols 16..31
        v2f b2 = *(const v2f*)(wp + 64);               // cols 32..47
        v2f b3 = *(const v2f*)(wp + 96);               // cols 48..63

        c0 = __builtin_amdgcn_wmma_f32_16x16x4_f32(false, a, false, b0,
                                                   (short)0, c0, false, false);
        c1 = __builtin_amdgcn_wmma_f32_16x16x4_f32(false, a, false, b1,
                                                   (short)0, c1, false, false);
        c2 = __builtin_amdgcn_wmma_f32_16x16x4_f32(false, a, false, b2,
                                                   (short)0, c2, false, false);
        c3 = __builtin_amdgcn_wmma_f32_16x16x4_f32(false, a, false, b3,
                                                   (short)0, c3, false, false);
    }

    float* yb = Y + (size_t)m0 * 64 + r;
#pragma unroll
    for (int j = 0; j < 8; ++j) {
        float* yr = yb + (size_t)(j + half * 8) * 64;
        yr[0]  = c0[j];
        yr[16] = c1[j];
        yr[32] = c2[j];
        yr[48] = c3[j];
    }
}

// alpha_src[n,h] = sum_c xh[n, h*C+c] * a_src[h,c]   (and same for dst)
__global__ void alpha_kernel(const float* __restrict__ xh,
                             const float* __restrict__ a_src,
                             const float* __restrict__ a_dst,
                             float* __restrict__ asrc,
                             float* __restrict__ adst,
                             int N, int H, int C)
{
    int idx = blockIdx.x * blockDim.x + threadIdx.x;
    if (idx >= N * H) return;
    int n = idx / H, h = idx % H;
    const float* row = xh + (size_t)n * 64 + h * C;
    const float* ws  = a_src + h * C;
    const float* wd  = a_dst + h * C;
    float s = 0.f, t = 0.f;
    for (int c = 0; c < C; ++c) {
        float v = row[c];
        s += v * ws[c];
        t += v * wd[c];
    }
    asrc[idx] = s;
    adst[idx] = t;
}

__device__ __forceinline__ void edge_sd(const int* __restrict__ ei,
                                        int e, int E, int& s, int& d) {
    if (e < E) { s = ei[e]; d = ei[E + e]; }
    else       { s = d = e - E; }          // self loop
}

__global__ void edge_max_kernel(const int* __restrict__ ei, int E, int Etot,
                                const float* __restrict__ asrc,
                                const float* __restrict__ adst,
                                float* __restrict__ emax, int H)
{
    long long idx = (long long)blockIdx.x * blockDim.x + threadIdx.x;
    if (idx >= (long long)Etot * H) return;
    int e = (int)(idx / H), h = (int)(idx % H);
    int s, d; edge_sd(ei, e, E, s, d);
    float v = lrelu02(asrc[(size_t)s * H + h] + adst[(size_t)d * H + h]);
    atomicMaxF(&emax[(size_t)d * H + h], v);
}

__global__ void edge_expsum_kernel(const int* __restrict__ ei, int E, int Etot,
                                   const float* __restrict__ asrc,
                                   const float* __restrict__ adst,
                                   const float* __restrict__ emax,
                                   float* __restrict__ denom, int H)
{
    long long idx = (long long)blockIdx.x * blockDim.x + threadIdx.x;
    if (idx >= (long long)Etot * H) return;
    int e = (int)(idx / H), h = (int)(idx % H);
    int s, d; edge_sd(ei, e, E, s, d);
    float v = lrelu02(asrc[(size_t)s * H + h] + adst[(size_t)d * H + h]);
    atomicAdd(&denom[(size_t)d * H + h], expf(v - emax[(size_t)d * H + h]));
}

// 8 threads per edge; each handles 8 consecutive channels of the 64-wide row.
__global__ void edge_agg_kernel(const int* __restrict__ ei, int E, int Etot,
                                const float* __restrict__ asrc,
                                const float* __restrict__ adst,
                                const float* __restrict__ emax,
                                const float* __restrict__ denom,
                                const float* __restrict__ xh,
                                float* __restrict__ agg, int H, int C)
{
    long long idx = (long long)blockIdx.x * blockDim.x + threadIdx.x;
    if (idx >= (long long)Etot * 8) return;
    int e = (int)(idx >> 3), g = (int)(idx & 7);
    int h = (g * 8) / C;                    // C=8 -> h=g ; C=64 -> h=0
    int s, d; edge_sd(ei, e, E, s, d);

    float logit = lrelu02(asrc[(size_t)s * H + h] + adst[(size_t)d * H + h]);
    float alpha = expf(logit - emax[(size_t)d * H + h]) /
                  (denom[(size_t)d * H + h] + 1e-16f);

    const float4* xr = (const float4*)(xh + (size_t)s * 64 + g * 8);
    float4 u = xr[0], w = xr[1];
    float* o = agg + (size_t)d * 64 + g * 8;
    atomicAdd(o + 0, u.x * alpha);
    atomicAdd(o + 1, u.y * alpha);
    atomicAdd(o + 2, u.z * alpha);
    atomicAdd(o + 3, u.w * alpha);
    atomicAdd(o + 4, w.x * alpha);
    atomicAdd(o + 5, w.y * alpha);
    atomicAdd(o + 6, w.z * alpha);
    atomicAdd(o + 7, w.w * alpha);
}

__global__ void bias_elu_kernel(float* __restrict__ p,
                                const float* __restrict__ b, size_t n)
{
    size_t i = (size_t)blockIdx.x * blockDim.x + threadIdx.x;
    if (i >= n) return;
    float v = p[i] + b[i & 63];
    p[i] = v > 0.f ? v : expm1f(v);
}

__global__ void bias_add_kernel(float* __restrict__ p,
                                const float* __restrict__ b, size_t n)
{
    size_t i = (size_t)blockIdx.x * blockDim.x + threadIdx.x;
    if (i >= n) return;
    p[i] = p[i] + b[i & 63];
}

// ---------------------------------------------------------------------------
static inline int nblk(long long n, int b) { return (int)((n + b - 1) / b); }

extern "C" void kernel_launch(void* const* d_in, const int* in_sizes, int n_in,
                              void* d_out, int out_size, void* d_ws, size_t ws_size,
                              hipStream_t stream)
{
    const int*   ei  = (const int*)  d_in[0];   // [2,E] int32
    const float* X   = (const float*)d_in[1];   // [N,128]
    const float* W1  = (const float*)d_in[2];   // [128,64]
    const float* as1 = (const float*)d_in[3];   // [8,8]
    const float* ad1 = (const float*)d_in[4];   // [8,8]
    const float* b1  = (const float*)d_in[5];   // [64]
    const float* W2  = (const float*)d_in[6];   // [64,64]
    const float* as2 = (const float*)d_in[7];   // [1,64]
    const float* ad2 = (const float*)d_in[8];   // [1,64]
    const float* b2  = (const float*)d_in[9];   // [64]

    const int E    = in_sizes[0] / 2;
    const int N    = in_sizes[1] / 128;
    const int Etot = E + N;                     // add_self_loops=True
    const size_t N64 = (size_t)N * 64;

    float* ws    = (float*)d_ws;
    float* bufA  = ws;                          // xh1, later y2     [N*64]
    float* bufB  = bufA + N64;                  // agg1 -> x2        [N*64]
    float* asrc1 = bufB + N64;                  // [N*8]
    float* adst1 = asrc1 + (size_t)N * 8;
    float* emax1 = adst1 + (size_t)N * 8;
    float* den1  = emax1 + (size_t)N * 8;
    float* asrc2 = den1  + (size_t)N * 8;       // [N]
    float* adst2 = asrc2 + N;
    float* emax2 = adst2 + N;
    float* den2  = emax2 + N;
    float* out   = (float*)d_out;               // [N,64]

    const int T = 256;

    // ---- per-call re-init (determinism) ----
    fill_kernel<<<nblk((long long)N * 8, T), T, 0, stream>>>(emax1, (size_t)N * 8, -INFINITY);
    fill_kernel<<<nblk((long long)N * 8, T), T, 0, stream>>>(den1,  (size_t)N * 8, 0.f);
    fill_kernel<<<nblk((long long)N64,   T), T, 0, stream>>>(bufB,  N64, 0.f);
    fill_kernel<<<nblk((long long)N,     T), T, 0, stream>>>(emax2, (size_t)N, -INFINITY);
    fill_kernel<<<nblk((long long)N,     T), T, 0, stream>>>(den2,  (size_t)N, 0.f);
    fill_kernel<<<nblk((long long)N64,   T), T, 0, stream>>>(out,   N64, 0.f);

    // ---- layer 1 : H=8, C=8 ----
    {
        const int H = 8, C = 8;
        int mtiles = (N + 15) / 16;
        gemm_wmma_kernel<<<nblk(mtiles, 8), 256, 0, stream>>>(X, W1, bufA, N, 128);
        alpha_kernel<<<nblk((long long)N * H, T), T, 0, stream>>>(bufA, as1, ad1, asrc1, adst1, N, H, C);
        edge_max_kernel<<<nblk((long long)Etot * H, T), T, 0, stream>>>(ei, E, Etot, asrc1, adst1, emax1, H);
        edge_expsum_kernel<<<nblk((long long)Etot * H, T), T, 0, stream>>>(ei, E, Etot, asrc1, adst1, emax1, den1, H);
        edge_agg_kernel<<<nblk((long long)Etot * 8, T), T, 0, stream>>>(ei, E, Etot, asrc1, adst1, emax1, den1, bufA, bufB, H, C);
        bias_elu_kernel<<<nblk((long long)N64, T), T, 0, stream>>>(bufB, b1, N64);
    }

    // ---- layer 2 : H=1, C=64 (mean over 1 head == identity) ----
    {
        const int H = 1, C = 64;
        int mtiles = (N + 15) / 16;
        gemm_wmma_kernel<<<nblk(mtiles, 8), 256, 0, stream>>>(bufB, W2, bufA, N, 64);
        alpha_kernel<<<nblk((long long)N * H, T), T, 0, stream>>>(bufA, as2, ad2, asrc2, adst2, N, H, C);
        edge_max_kernel<<<nblk((long long)Etot * H, T), T, 0, stream>>>(ei, E, Etot, asrc2, adst2, emax2, H);
        edge_expsum_kernel<<<nblk((long long)Etot * H, T), T, 0, stream>>>(ei, E, Etot, asrc2, adst2, emax2, den2, H);
        edge_agg_kernel<<<nblk((long long)Etot * 8, T), T, 0, stream>>>(ei, E, Etot, asrc2, adst2, emax2, den2, bufA, out, H, C);
        bias_add_kernel<<<nblk((long long)N64, T), T, 0, stream>>>(out, b2, N64);
    }
}